// ChebyshevActivation_8409545966397
// MI455X (gfx1250) — compile-verified
//
#include <hip/hip_runtime.h>

#define IN_FEATURES 2048
#define OUT_FEATURES 1024
#define BATCH 8192
#define WIDTH 9     // Chebyshev degrees 0..8
#define KPAD 12     // padded K for wmma f32 16x16x4 (3 k-steps of 4)

typedef __attribute__((ext_vector_type(2))) float v2f;
typedef __attribute__((ext_vector_type(4))) float v4f;
typedef __attribute__((ext_vector_type(8))) float v8f;

__device__ __forceinline__ float fast_tanh(float v) {
#if __has_builtin(__builtin_amdgcn_tanhf)
    return __builtin_amdgcn_tanhf(v);   // v_tanh_f32 (TRANS32)
#else
    return tanhf(v);
#endif
}

// -----------------------------------------------------------------------------
// Pass 0: zero-padded transpose of coeffs: CtPad[k][n] = coeffs[n][k], k<9
// 12 x 1024 floats, one thread per element.
// -----------------------------------------------------------------------------
__global__ __launch_bounds__(256) void pad_coeffs_kernel(
    const float* __restrict__ coeffs, float* __restrict__ CtPad)
{
    const int tid = blockIdx.x * 256 + threadIdx.x;   // 0 .. 12*1024-1
    const int k = tid >> 10;
    const int n = tid & 1023;
    CtPad[tid] = (k < WIDTH) ? coeffs[n * WIDTH + k] : 0.0f;
}

// -----------------------------------------------------------------------------
// Pass 1: S[b][w] = sum_i T_w(tanh(x[b][i] * scale)), w = 0..8, padded to 12.
// One wave32 per batch row; 2048 / (32 lanes * 4) = 16 float4 loads per lane.
// Inner math is expressed on <2 x float> vectors so the backend can select
// V_PK_FMA_F32 / V_PK_ADD_F32 (VOP3P packed f32), halving VALU slots for the
// Chebyshev recursion + accumulation. tanh stays scalar TRANS32 (co-executes).
// The reference's clip(+-100) is the identity (tanh in [-1,1] implies
// |2 t T_cur - T_prev| <= 3), so it is omitted bit-safely.
// -----------------------------------------------------------------------------
__global__ __launch_bounds__(256) void cheb_reduce_kernel(
    const float* __restrict__ x,
    const float* __restrict__ input_scale,
    float* __restrict__ S)
{
    const int lane = threadIdx.x & 31;
    const int wave = threadIdx.x >> 5;
    const int row  = blockIdx.x * 8 + wave;

    float scale = input_scale[0];
    scale = fminf(fmaxf(scale, 0.1f), 2.0f);

    const float* xr = x + (size_t)row * IN_FEATURES;

    v2f s1 = {0.f, 0.f}, s2 = {0.f, 0.f}, s3 = {0.f, 0.f}, s4 = {0.f, 0.f},
        s5 = {0.f, 0.f}, s6 = {0.f, 0.f}, s7 = {0.f, 0.f}, s8 = {0.f, 0.f};

    #pragma unroll
    for (int j = 0; j < IN_FEATURES / (32 * 4); ++j) {
        v4f xv = __builtin_nontemporal_load(
                     (const v4f*)(xr + j * 128 + lane * 4));
        #pragma unroll
        for (int h = 0; h < 2; ++h) {      // two <2 x float> element pairs
            v2f t;
            t.x = fast_tanh(xv[2 * h + 0] * scale);
            t.y = fast_tanh(xv[2 * h + 1] * scale);
            v2f t2 = t + t;
            // T0 = 1, T1 = t, T_n = 2*t*T_{n-1} - T_{n-2}   (packed pairs)
            v2f T2 = t2 * t  - 1.0f;
            v2f T3 = t2 * T2 - t;
            v2f T4 = t2 * T3 - T2;
            v2f T5 = t2 * T4 - T3;
            v2f T6 = t2 * T5 - T4;
            v2f T7 = t2 * T6 - T5;
            v2f T8 = t2 * T7 - T6;
            s1 += t;  s2 += T2; s3 += T3; s4 += T4;
            s5 += T5; s6 += T6; s7 += T7; s8 += T8;
        }
    }

    // collapse packed pairs, then wave32 butterfly reduction
    float r1 = s1.x + s1.y, r2 = s2.x + s2.y, r3 = s3.x + s3.y,
          r4 = s4.x + s4.y, r5 = s5.x + s5.y, r6 = s6.x + s6.y,
          r7 = s7.x + s7.y, r8 = s8.x + s8.y;

    #pragma unroll
    for (int m = 16; m >= 1; m >>= 1) {
        r1 += __shfl_xor(r1, m, 32);
        r2 += __shfl_xor(r2, m, 32);
        r3 += __shfl_xor(r3, m, 32);
        r4 += __shfl_xor(r4, m, 32);
        r5 += __shfl_xor(r5, m, 32);
        r6 += __shfl_xor(r6, m, 32);
        r7 += __shfl_xor(r7, m, 32);
        r8 += __shfl_xor(r8, m, 32);
    }

    if (lane == 0) {
        float* Sr = S + (size_t)row * KPAD;
        Sr[0]  = (float)IN_FEATURES;  // sum of T0 == 1 over 2048 elements
        Sr[1]  = r1;  Sr[2] = r2;  Sr[3] = r3;  Sr[4] = r4;
        Sr[5]  = r5;  Sr[6] = r6;  Sr[7] = r7;  Sr[8] = r8;
        Sr[9]  = 0.f; Sr[10] = 0.f; Sr[11] = 0.f;  // K padding (zeros)
    }
}

// -----------------------------------------------------------------------------
// Pass 2: out = S (8192x12) @ CtPad (12x1024) via V_WMMA_F32_16X16X4_F32,
// 3 k-steps of 4. One wave per 16x16 output tile; 512 x 64 tiles, 8 waves/blk.
// No branches: EXEC is all-ones at every WMMA.
//
// A (16x4 f32): lane M = lane&15; vgpr0/1 hold K = 2*(lane>>4) + {0,1}.
// B (4x16 f32): lane N = lane&15; vgpr0/1 hold K = 2*(lane>>4) + {0,1}.
// C/D (16x16):  vgpr r -> row r (lanes 0-15) / row r+8 (lanes 16-31), col lane&15.
// -----------------------------------------------------------------------------
__global__ __launch_bounds__(256) void cheb_wmma_kernel(
    const float* __restrict__ S,
    const float* __restrict__ CtPad,
    float* __restrict__ out)
{
    const int lane = threadIdx.x & 31;
    const int wave = threadIdx.x >> 5;
    const int tile = blockIdx.x * 8 + wave;
    const int rowTile = tile >> 6;   // / 64 col-tiles
    const int colTile = tile & 63;
    const int b0   = rowTile * 16;
    const int n0   = colTile * 16;
    const int half = lane >> 4;      // 0: lanes 0-15, 1: lanes 16-31
    const int l    = lane & 15;
    const int n    = n0 + l;

    v8f c = {};

    #pragma unroll
    for (int kk = 0; kk < 3; ++kk) {
        const int kbase = kk * 4 + half * 2;

        // 8-byte aligned: (row*12 + even_k)*4 bytes
        v2f a = *(const v2f*)(S + (size_t)(b0 + l) * KPAD + kbase);

        v2f b;
        b.x = CtPad[(kbase    ) * OUT_FEATURES + n];
        b.y = CtPad[(kbase + 1) * OUT_FEATURES + n];

        c = __builtin_amdgcn_wmma_f32_16x16x4_f32(
                /*neg_a=*/false, a, /*neg_b=*/false, b,
                /*c_mod=*/(short)0, c, /*reuse_a=*/false, /*reuse_b=*/false);
    }

    #pragma unroll
    for (int r = 0; r < 8; ++r) {
        __builtin_nontemporal_store(
            c[r],
            out + (size_t)(b0 + r + half * 8) * OUT_FEATURES + n);
    }
}

extern "C" void kernel_launch(void* const* d_in, const int* in_sizes, int n_in,
                              void* d_out, int out_size, void* d_ws, size_t ws_size,
                              hipStream_t stream) {
    const float* x           = (const float*)d_in[0];   // (8192, 2048) f32
    const float* coeffs      = (const float*)d_in[1];   // (1024, 9)    f32
    const float* input_scale = (const float*)d_in[2];   // (1,)         f32
    float* out = (float*)d_out;                          // (8192, 1024) f32

    // workspace layout: S (8192*12 f32 = 384 KB) | CtPad (12*1024 f32 = 48 KB)
    float* S     = (float*)d_ws;
    float* CtPad = S + (size_t)BATCH * KPAD;

    // Pass 0: zero-padded coeffs^T (12 x 1024)
    pad_coeffs_kernel<<<(KPAD * OUT_FEATURES) / 256, 256, 0, stream>>>(coeffs, CtPad);

    // Pass 1: per-row Chebyshev sums -> S (zero-padded K=12)
    cheb_reduce_kernel<<<BATCH / 8, 256, 0, stream>>>(x, input_scale, S);

    // Pass 2: out = S @ coeffs^T via f32 WMMA
    const int tiles = (BATCH / 16) * (OUT_FEATURES / 16); // 32768
    cheb_wmma_kernel<<<tiles / 8, 256, 0, stream>>>(S, CtPad, out);
}